// qlayer_40243843564158
// MI455X (gfx1250) — compile-verified
//
#include <hip/hip_runtime.h>
#include <math.h>

// ---------------------------------------------------------------------------
// Quantum-layer style pipeline for MI455X (gfx1250, wave32, WMMA):
//   out = |((normalize(x) @ M1^T) * e^{i phi}) @ M2^T|^2
// M1/M2 real 256x256 (Givens products) -> 3 real fp32 GEMMs via
// V_WMMA_F32_16X16X4_F32, fully fused per 64-row batch tile.
// ---------------------------------------------------------------------------

#define NDIM   256
#define NBATCH 65536
#define NROT   32640          // 256*255/2
#define TILE_B 64             // batch rows per workgroup
#define RS     260            // padded LDS row stride (floats): banks (4m+k)%64

typedef __attribute__((ext_vector_type(2))) float v2f;
typedef __attribute__((ext_vector_type(8))) float v8f;

// ---------------------------------------------------------------------------
// Kernel 0: precompute cos/sin of all rotation angles and phases (parallel).
// ---------------------------------------------------------------------------
__global__ void qlayer_trig(const float* __restrict__ rots1,
                            const float* __restrict__ rots2,
                            const float* __restrict__ phases,
                            float* __restrict__ c1, float* __restrict__ s1,
                            float* __restrict__ c2, float* __restrict__ s2,
                            float* __restrict__ cp, float* __restrict__ sp) {
  int idx = blockIdx.x * blockDim.x + threadIdx.x;
  if (idx < NROT) {
    float s, c;
    sincosf(rots1[idx], &s, &c); c1[idx] = c; s1[idx] = s;
    sincosf(rots2[idx], &s, &c); c2[idx] = c; s2[idx] = s;
  }
  if (idx < NDIM) {
    float s, c;
    sincosf(phases[idx], &s, &c); cp[idx] = c; sp[idx] = s;
  }
}

// ---------------------------------------------------------------------------
// Kernel 1: build M = G_last ... G_1 (identity seeded). Left row-rotations act
// on each COLUMN independently -> thread t evolves column t (e_t) through all
// 32640 2x2 rotations with its private 256-float vector in LDS (stride 257 ->
// conflict-free across the wave). No barriers needed. Output is stored
// K-major: T[k*256 + n] = M[n][k], which is exactly the B-operand layout the
// GEMMs want (coalesced along n).
// ---------------------------------------------------------------------------
__global__ void qlayer_givens(const float* __restrict__ c1,
                              const float* __restrict__ s1,
                              const float* __restrict__ c2,
                              const float* __restrict__ s2,
                              float* __restrict__ T1,
                              float* __restrict__ T2) {
  extern __shared__ float sm[];          // 256 threads * 257 floats = 263168 B
  const int t = threadIdx.x;
  const float* cA = (blockIdx.x == 0) ? c1 : c2;
  const float* sA = (blockIdx.x == 0) ? s1 : s2;
  float* T        = (blockIdx.x == 0) ? T1 : T2;

  float* v = sm + t * 257;
  for (int k = 0; k < NDIM; ++k) v[k] = (k == t) ? 1.0f : 0.0f;

  int i = 0, j = 1;                      // combinations(range(256), 2) order
  for (int r = 0; r < NROT; ++r) {
    float c = cA[r];
    float s = sA[r];
    float ri = v[i];
    float rj = v[j];
    v[i] = c * ri - s * rj;
    v[j] = s * ri + c * rj;
    if (++j == NDIM) { ++i; j = i + 1; }
  }
  // thread t holds column t of M: v[n] = M[n][t]  ->  T[t*256 + n] = M[n][t]
  for (int n = 0; n < NDIM; ++n) T[t * NDIM + n] = v[n];
}

// ---------------------------------------------------------------------------
// GEMM helper: accumulate a 16x128 output stripe (8 WMMA tiles) for this wave.
//   D[m][n] += sum_k A[m][k] * M[n][k],  with T[k*256+n] = M[n][k].
// A-fragment (16x4 fp32): lanes 0-15 hold M=lane, K=k,k+1; lanes 16-31 hold
// M=lane-16, K=k+2,k+3.  B-fragment mirrors along K.
// ---------------------------------------------------------------------------
__device__ __forceinline__ void gemm_stripe(const float* __restrict__ smSrc,
                                            const float* __restrict__ T,
                                            int r, int h, int ln16, int hl,
                                            v8f acc[8]) {
  const int m = r * 16 + ln16;           // A row for this lane
  #pragma unroll 2
  for (int k = 0; k < NDIM; k += 4) {
    const int kk = k + 2 * hl;
    v2f a;
    a.x = smSrc[m * RS + kk];
    a.y = smSrc[m * RS + kk + 1];
    const float* b0 = T + kk * NDIM       + h * 128 + ln16;
    const float* b1 = T + (kk + 1) * NDIM + h * 128 + ln16;
    #pragma unroll
    for (int t = 0; t < 8; ++t) {
      v2f b;
      b.x = b0[t * 16];
      b.y = b1[t * 16];
      // D = A(16x4) x B(4x16) + C   (fp32 WMMA, 8-arg form)
      acc[t] = __builtin_amdgcn_wmma_f32_16x16x4_f32(
          /*neg_a=*/false, a, /*neg_b=*/false, b,
          /*c_mod=*/(short)0, acc[t], /*reuse_a=*/false, /*reuse_b=*/false);
    }
  }
}

// ---------------------------------------------------------------------------
// Kernel 2: fused normalize -> GEMM1 -> phases -> 2x GEMM2 -> |.|^2.
// 256 threads (8 waves). Wave w: row-tile r = w&3 (16 rows), col-half h = w>>2
// (8 x 16-col tiles). LDS: smA (Xn, later YR) and smB (scratch, later YI),
// each 64 x 260 floats.
// ---------------------------------------------------------------------------
__global__ void __launch_bounds__(256)
qlayer_main(const float* __restrict__ x,
            const float* __restrict__ T1,
            const float* __restrict__ T2,
            const float* __restrict__ cp,
            const float* __restrict__ sp,
            float* __restrict__ out) {
  extern __shared__ float sm[];
  float* smA = sm;                   // TILE_B * RS floats
  float* smB = sm + TILE_B * RS;     // TILE_B * RS floats

  const int tid  = threadIdx.x;
  const int wave = tid >> 5;
  const int lane = tid & 31;
  const int hl   = lane >> 4;        // 0: lanes 0-15, 1: lanes 16-31
  const int ln16 = lane & 15;
  const int r    = wave & 3;         // row-tile within 64-row batch tile
  const int h    = wave >> 2;        // column half (128 cols)
  const int base_row = blockIdx.x * TILE_B;

  // ---- load 64x256 x-tile, row-normalize into smA -------------------------
  {
    const int row = tid >> 2;        // 64 rows, 4 threads each
    const int q   = tid & 3;         // 64-col quarter
    const float4* src = (const float4*)(x + (base_row + row) * NDIM + q * 64);
    float4* dst = (float4*)(smA + row * RS + q * 64);   // RS%4==0 -> aligned
    float acc = 0.f;
    #pragma unroll
    for (int c = 0; c < 16; ++c) {
      float4 vv = src[c];
      dst[c] = vv;
      acc += vv.x * vv.x + vv.y * vv.y + vv.z * vv.z + vv.w * vv.w;
    }
    smB[row * 4 + q] = acc;
  }
  __syncthreads();
  if (tid < TILE_B) {
    float s = smB[tid * 4] + smB[tid * 4 + 1] + smB[tid * 4 + 2] + smB[tid * 4 + 3];
    smB[1024 + tid] = rsqrtf(s);
  }
  __syncthreads();
  {
    const int row = tid >> 2;
    const int q   = tid & 3;
    const float inv = smB[1024 + row];
    float4* dst = (float4*)(smA + row * RS + q * 64);
    #pragma unroll
    for (int c = 0; c < 16; ++c) {
      float4 vv = dst[c];
      vv.x *= inv; vv.y *= inv; vv.z *= inv; vv.w *= inv;
      dst[c] = vv;
    }
  }
  __syncthreads();

  const v8f vzero = {0.f, 0.f, 0.f, 0.f, 0.f, 0.f, 0.f, 0.f};

  // ---- GEMM1: Y1 = Xn @ M1^T ---------------------------------------------
  v8f acc[8];
  #pragma unroll
  for (int t = 0; t < 8; ++t) acc[t] = vzero;
  gemm_stripe(smA, T1, r, h, ln16, hl, acc);

  __syncthreads();                   // all waves done reading smA (Xn)

  // ---- phases: YR = Y1*cos(phi_n), YI = Y1*sin(phi_n) ---------------------
  // C/D layout: VGPR g -> M = g + 8*hl, N = ln16 (per 16-col tile).
  #pragma unroll
  for (int t = 0; t < 8; ++t) {
    const int n  = h * 128 + t * 16 + ln16;
    const float cw = cp[n];
    const float sw = sp[n];
    #pragma unroll
    for (int g = 0; g < 8; ++g) {
      const int m = r * 16 + g + 8 * hl;
      const float y = acc[t][g];
      smA[m * RS + n] = y * cw;      // YR
      smB[m * RS + n] = y * sw;      // YI
    }
  }
  __syncthreads();

  // ---- GEMM2 (real): out += (YR @ M2^T)^2 ---------------------------------
  v8f res[8];
  #pragma unroll
  for (int t = 0; t < 8; ++t) acc[t] = vzero;
  gemm_stripe(smA, T2, r, h, ln16, hl, acc);
  #pragma unroll
  for (int t = 0; t < 8; ++t) {
    #pragma unroll
    for (int g = 0; g < 8; ++g) { float v = acc[t][g]; res[t][g] = v * v; }
  }

  // ---- GEMM2 (imag): out += (YI @ M2^T)^2 ---------------------------------
  #pragma unroll
  for (int t = 0; t < 8; ++t) acc[t] = vzero;
  gemm_stripe(smB, T2, r, h, ln16, hl, acc);
  #pragma unroll
  for (int t = 0; t < 8; ++t) {
    #pragma unroll
    for (int g = 0; g < 8; ++g) { float v = acc[t][g]; res[t][g] += v * v; }
  }

  // ---- store |y|^2 --------------------------------------------------------
  #pragma unroll
  for (int t = 0; t < 8; ++t) {
    const int n = h * 128 + t * 16 + ln16;
    #pragma unroll
    for (int g = 0; g < 8; ++g) {
      const int m = r * 16 + g + 8 * hl;
      out[(base_row + m) * NDIM + n] = res[t][g];
    }
  }
}

// ---------------------------------------------------------------------------
// Host launcher. Inputs (setup_inputs order): x, rots1, phases, rots2.
// Workspace layout (floats):
//   c1[32640] s1[32640] c2[32640] s2[32640] cp[256] sp[256]
//   T1[65536] T2[65536]                       (total 1 MiB)
// ---------------------------------------------------------------------------
extern "C" void kernel_launch(void* const* d_in, const int* in_sizes, int n_in,
                              void* d_out, int out_size, void* d_ws, size_t ws_size,
                              hipStream_t stream) {
  const float* x      = (const float*)d_in[0];
  const float* rots1  = (const float*)d_in[1];
  const float* phases = (const float*)d_in[2];
  const float* rots2  = (const float*)d_in[3];
  float* out = (float*)d_out;
  float* ws  = (float*)d_ws;

  float* c1 = ws;
  float* s1 = ws + 32640;
  float* c2 = ws + 65280;
  float* s2 = ws + 97920;
  float* cp = ws + 130560;
  float* sp = ws + 130816;
  float* T1 = ws + 131072;
  float* T2 = ws + 196608;

  const int lds_givens = 256 * 257 * 4;          // 263168 B (<= 320 KB/WGP)
  const int lds_main   = 2 * TILE_B * RS * 4;    // 133120 B (2 WG/WGP fits)
  hipFuncSetAttribute(reinterpret_cast<const void*>(qlayer_givens),
                      hipFuncAttributeMaxDynamicSharedMemorySize, lds_givens);
  hipFuncSetAttribute(reinterpret_cast<const void*>(qlayer_main),
                      hipFuncAttributeMaxDynamicSharedMemorySize, lds_main);

  qlayer_trig<<<(NROT + 255) / 256, 256, 0, stream>>>(
      rots1, rots2, phases, c1, s1, c2, s2, cp, sp);
  qlayer_givens<<<2, 256, lds_givens, stream>>>(c1, s1, c2, s2, T1, T2);
  qlayer_main<<<NBATCH / TILE_B, 256, lds_main, stream>>>(
      x, T1, T2, cp, sp, out);
}